// OlmoeSparseMoeBlock_63763084476935
// MI455X (gfx1250) — compile-verified
//
#include <hip/hip_runtime.h>
#include <hip/hip_bf16.h>

// ---------------------------------------------------------------------------
// OLMoE sparse MoE block for MI455X (gfx1250), fp32 via V_WMMA_F32_16X16X4_F32
// Memory-bound design: every expert weight byte streamed exactly once.
// X tiles staged to LDS via CDNA5 async-to-LDS (ASYNCcnt) when available.
// ---------------------------------------------------------------------------

typedef __attribute__((ext_vector_type(2))) float v2f;
typedef __attribute__((ext_vector_type(8))) float v8f;

// exact pointee type expected by __builtin_amdgcn_global_load_async_to_lds_b128
typedef int b128_t __attribute__((vector_size(4 * sizeof(int))));
typedef __attribute__((address_space(1))) b128_t* g_b128p;   // global (device)
typedef __attribute__((address_space(3))) b128_t* l_b128p;   // LDS (shared)

#define HIDDEN   2048
#define INTER    1024
#define NEXP     64
#define TOPK     8
#define NTOK     512

#define SLICES   8
#define SLICE_I  (INTER / SLICES)     // 128 inter columns per block
#define KC       512                  // K-chunk of X staged in LDS
#define XS_STRIDE (KC + 4)            // 516 floats: 516 % 64 == 4 -> conflict-free b64
#define AS_STRIDE (SLICE_I + 4)       // 132 floats: 132 % 64 == 4

#if defined(__HIP_DEVICE_COMPILE__) && __has_builtin(__builtin_amdgcn_global_load_async_to_lds_b128)
#define HAVE_ASYNC_LDS 1
#else
#define HAVE_ASYNC_LDS 0
#endif

// ---------------------------------------------------------------------------
// Kernel 0: zero output + per-expert counters (output is atomically built)
// ---------------------------------------------------------------------------
__global__ void moe_zero_kernel(float* __restrict__ out, int n, int* __restrict__ counts) {
    int i = blockIdx.x * blockDim.x + threadIdx.x;
    if (i < n) out[i] = 0.0f;
    if (i < NEXP) counts[i] = 0;
}

// ---------------------------------------------------------------------------
// Kernel 1: router. One block per token (64 threads = 2 waves).
// logits[e] = dot(x[t], w_router[:, e]); top-8 + softmax; append to expert
// token lists with atomics. Tiny vs. expert compute, simplicity wins.
// ---------------------------------------------------------------------------
__global__ __launch_bounds__(64)
void moe_router_kernel(const float* __restrict__ x,
                       const float* __restrict__ w_router,
                       int*   __restrict__ counts,
                       int*   __restrict__ tok_idx,
                       float* __restrict__ tok_gate) {
    __shared__ float lg[NEXP];
    const int t = blockIdx.x;
    const int e = threadIdx.x;

    const float* xr = x + (size_t)t * HIDDEN;
    float s = 0.0f;
    #pragma unroll 8
    for (int d = 0; d < HIDDEN; ++d)
        s += xr[d] * w_router[(size_t)d * NEXP + e];   // coalesced over e
    lg[e] = s;
    __syncthreads();

    if (e == 0) {
        float sc[TOPK];
        int   id[TOPK];
        for (int k = 0; k < TOPK; ++k) {
            float best = -3.0e38f; int bi = 0;
            for (int j = 0; j < NEXP; ++j)
                if (lg[j] > best) { best = lg[j]; bi = j; }
            sc[k] = best; id[k] = bi; lg[bi] = -3.0e38f;
        }
        // softmax over the (descending) top-8; sc[0] is the max
        float mx = sc[0], sum = 0.0f, g[TOPK];
        for (int k = 0; k < TOPK; ++k) { g[k] = __expf(sc[k] - mx); sum += g[k]; }
        float inv = 1.0f / sum;
        for (int k = 0; k < TOPK; ++k) {
            int p = atomicAdd(&counts[id[k]], 1);
            tok_idx [id[k] * NTOK + p] = t;
            tok_gate[id[k] * NTOK + p] = g[k] * inv;
        }
    }
}

// ---------------------------------------------------------------------------
// Kernel 2: expert compute. grid = (SLICES, NEXP) = 512 blocks, 8 waves each.
// Block (s, e) owns INTER columns [s*128, s*128+128) of expert e:
//   h,u = X_e @ Wg[:,slice], X_e @ Wu[:,slice]    (WMMA f32 16x16x4)
//   a   = silu(h) * u * gate                       (fused, staged in LDS)
//   out += a @ Wd[slice,:]                         (WMMA + global f32 atomics)
// Weight reads are disjoint across slices => each weight byte read once.
// One gate-tile + one up-tile per wave => low VGPR pressure, deep occupancy
// for latency hiding of the weight stream (the binding resource is HBM BW).
// ---------------------------------------------------------------------------
__global__ __launch_bounds__(256)
void moe_expert_kernel(const float* __restrict__ x,
                       const float* __restrict__ w_gate,
                       const float* __restrict__ w_up,
                       const float* __restrict__ w_down,
                       const int*   __restrict__ counts,
                       const int*   __restrict__ tok_idx,
                       const float* __restrict__ tok_gate,
                       float*       __restrict__ out) {
    __shared__ float xs[16 * XS_STRIDE];   // 33.0 KB gathered X K-chunk
    __shared__ float as[16 * AS_STRIDE];   //  8.5 KB activated tile
    __shared__ int   sTok[16];
    __shared__ float sGate[16];

    const int slice = blockIdx.x;
    const int e     = blockIdx.y;
    const int i0    = slice * SLICE_I;

    const int tid  = threadIdx.x;
    const int wave = tid >> 5;
    const int lane = tid & 31;
    const int half = lane >> 4;   // A/B fragment half: K += 2*half
    const int l16  = lane & 15;

    const int cnt = counts[e];

    // per-wave gate/up output column (one 16-wide N-tile per wave)
    const int n0 = i0 + wave * 16 + l16;

    const size_t ebaseH = (size_t)e * HIDDEN;   // row base into [E, HIDDEN, INTER]
    const size_t ebaseI = (size_t)e * INTER;    // row base into [E, INTER, HIDDEN]

    for (int m0 = 0; m0 < cnt; m0 += 16) {
        __syncthreads();  // previous M-tile fully consumed (incl. sTok readers)
        if (tid < 16) {
            int j = m0 + tid;
            sTok [tid] = (j < cnt) ? tok_idx [e * NTOK + j] : 0;
            sGate[tid] = (j < cnt) ? tok_gate[e * NTOK + j] : 0.0f;
        }

        v8f acc_g = {}, acc_u = {};

        for (int kc = 0; kc < HIDDEN; kc += KC) {
            __syncthreads();  // sTok ready / previous chunk consumed
            // gather 16 token rows x KC floats, 16B per lane per transfer
            for (int i = tid; i < 16 * (KC / 4); i += 256) {
                int r  = i >> 7;          // KC/4 == 128
                int c4 = i & 127;
                const float* src = x + (size_t)sTok[r] * HIDDEN + kc + c4 * 4;
                float*       dst = &xs[r * XS_STRIDE + c4 * 4];
#if HAVE_ASYNC_LDS
                // CDNA5 async copy: memory -> LDS without touching VGPRs
                __builtin_amdgcn_global_load_async_to_lds_b128(
                    (g_b128p)(void*)src, (l_b128p)(void*)dst, 0, 0);
#else
                *reinterpret_cast<float4*>(dst) =
                    *reinterpret_cast<const float4*>(src);
#endif
            }
            // speculative warm of next weight K-chunk (global_prefetch_b8)
            if (kc + KC < HIDDEN) {
                size_t row = ebaseH + (size_t)(kc + KC) + (tid >> 2);
                int    col = i0 + (tid & 3) * 32;
                __builtin_prefetch(w_gate + row * INTER + col, 0, 0);
                __builtin_prefetch(w_up   + row * INTER + col, 0, 0);
            }
#if HAVE_ASYNC_LDS
#if __has_builtin(__builtin_amdgcn_s_wait_asynccnt)
            __builtin_amdgcn_s_wait_asynccnt(0);
#else
            asm volatile("s_wait_asynccnt 0" ::: "memory");
#endif
#endif
            __syncthreads();

            // streaming B pointers; advance 4 K-rows per WMMA step
            const float* pg = w_gate + (ebaseH + kc + 2 * half) * INTER + n0;
            const float* pu = w_up   + (ebaseH + kc + 2 * half) * INTER + n0;

            #pragma unroll 4
            for (int k = 0; k < KC; k += 4) {
                const float* xr = &xs[l16 * XS_STRIDE + k + 2 * half];
                v2f a = { xr[0], xr[1] };                 // ds_load_b64, conflict-free
                v2f bg = { pg[0], pg[INTER] };
                v2f bu = { pu[0], pu[INTER] };
                acc_g = __builtin_amdgcn_wmma_f32_16x16x4_f32(false, a, false, bg, (short)0, acc_g, false, false);
                acc_u = __builtin_amdgcn_wmma_f32_16x16x4_f32(false, a, false, bu, (short)0, acc_u, false, false);
                pg += (size_t)4 * INTER;
                pu += (size_t)4 * INTER;
            }
        }

        __syncthreads();
        // fuse: a = silu(h) * u * gate  -> LDS tile for the down projection
        #pragma unroll
        for (int r = 0; r < 8; ++r) {
            int m = r + 8 * half;                         // C/D row mapping
            float h = acc_g[r];
            float a = (h / (1.0f + __expf(-h))) * acc_u[r] * sGate[m];
            as[m * AS_STRIDE + wave * 16 + l16] = a;
        }
        __syncthreads();

        // down projection: y(16 x 2048) = a(16 x 128) @ Wd[slice](128 x 2048)
        for (int dt = wave; dt < HIDDEN / 16; dt += 8) {
            int d = dt * 16 + l16;
            const float* pd = w_down + (ebaseI + i0 + 2 * half) * HIDDEN + d;
            v8f c = {};
            #pragma unroll 4
            for (int k = 0; k < SLICE_I; k += 4) {
                const float* ar = &as[l16 * AS_STRIDE + k + 2 * half];
                v2f a = { ar[0], ar[1] };
                v2f b = { pd[0], pd[HIDDEN] };
                c = __builtin_amdgcn_wmma_f32_16x16x4_f32(false, a, false, b, (short)0, c, false, false);
                pd += (size_t)4 * HIDDEN;
            }
            #pragma unroll
            for (int r = 0; r < 8; ++r) {
                int m = r + 8 * half;
                if (m0 + m < cnt)
                    atomicAdd(out + (size_t)sTok[m] * HIDDEN + d, c[r]);
            }
        }
        __syncthreads();  // all waves done with sTok/as before next M-tile
    }
}

// ---------------------------------------------------------------------------
// Launch
// ---------------------------------------------------------------------------
extern "C" void kernel_launch(void* const* d_in, const int* in_sizes, int n_in,
                              void* d_out, int out_size, void* d_ws, size_t ws_size,
                              hipStream_t stream) {
    (void)in_sizes; (void)n_in; (void)out_size; (void)ws_size;

    const float* x        = (const float*)d_in[0];  // [1, 512, 2048]
    const float* w_router = (const float*)d_in[1];  // [2048, 64]
    const float* w_gate   = (const float*)d_in[2];  // [64, 2048, 1024]
    const float* w_up     = (const float*)d_in[3];  // [64, 2048, 1024]
    const float* w_down   = (const float*)d_in[4];  // [64, 1024, 2048]
    float* out = (float*)d_out;                     // [1, 512, 2048]

    // workspace layout: counts | tok_idx | tok_gate   (~262 KB)
    int*   counts   = (int*)d_ws;
    int*   tok_idx  = (int*)((char*)d_ws + 256);
    float* tok_gate = (float*)((char*)d_ws + 256 + NEXP * NTOK * sizeof(int));

    const int n_out = NTOK * HIDDEN;
    moe_zero_kernel<<<(n_out + 255) / 256, 256, 0, stream>>>(out, n_out, counts);
    moe_router_kernel<<<NTOK, 64, 0, stream>>>(x, w_router, counts, tok_idx, tok_gate);
    moe_expert_kernel<<<dim3(SLICES, NEXP), 256, 0, stream>>>(
        x, w_gate, w_up, w_down, counts, tok_idx, tok_gate, out);
}